// Loss_90640989815440
// MI455X (gfx1250) — compile-verified
//
#include <hip/hip_runtime.h>
#include <cstdint>

// ---------------- problem constants (from reference) ----------------
#define B_      16
#define C_      80
#define S_      17064
#define ALPHA_  0.25f
#define CSPLIT  4                 // channel chunks per block (more parallelism)
#define CCHUNK  (C_ / CSPLIT)     // 20 channels per block
#define TILE_J  1024              // 256 lanes * 4 contiguous j each

// ---------------- CDNA5 feature detection (compile-loop guarded) ----------------
#if defined(__HIP_DEVICE_COMPILE__) && defined(__has_builtin)
#  if __has_builtin(__builtin_amdgcn_global_load_async_to_lds_b128)
#    define HAVE_ASYNC_LDS 1
#  endif
#  if __has_builtin(__builtin_amdgcn_wmma_f32_16x16x4_f32)
#    define HAVE_WMMA_X4 1
#  endif
#endif

#ifdef HAVE_ASYNC_LDS
#  if __has_builtin(__builtin_amdgcn_s_wait_asynccnt)
#    define WAIT_ASYNC(n) do { __builtin_amdgcn_s_wait_asynccnt(n); asm volatile("" ::: "memory"); } while (0)
#  else
#    define WAIT_ASYNC(n) asm volatile("s_wait_asynccnt %0" :: "i"(n) : "memory")
#  endif
// builtin signature (from diagnostic): b128 variant takes pointers to 128-bit int
// vectors; global side in addrspace(1) ("__device__"), LDS side in addrspace(3).
typedef int v4i_ __attribute__((vector_size(16)));
typedef __attribute__((address_space(1))) v4i_* glob_v4i_p;
typedef __attribute__((address_space(3))) v4i_* lds_v4i_p;
__device__ __forceinline__ void async_b128(const float* g, float* l) {
  // global_load_async_to_lds_b128: per-lane 16B, tracked by ASYNCcnt
  __builtin_amdgcn_global_load_async_to_lds_b128((glob_v4i_p)g, (lds_v4i_p)l, 0, 0);
}
#endif

typedef __attribute__((ext_vector_type(2))) float v2f;
typedef __attribute__((ext_vector_type(8))) float v8f;

__device__ __forceinline__ float frcp(float x) {
#if defined(__HIP_DEVICE_COMPILE__) && __has_builtin(__builtin_amdgcn_rcpf)
  return __builtin_amdgcn_rcpf(x);           // v_rcp_f32
#else
  return 1.0f / x;
#endif
}

// ---- wave32 all-lane sum. WMMA ones-matrix trick:
// A(16x4 f32): lane m in 0..15 -> {A[m,0],A[m,1]} = {p,0}; lane m+16 -> {A[m,2],A[m,3]} = {p,0}
// B = ones => D[m,n] = p_m + p_{m+16}. Summing one lane's 8 D-VGPRs gives a half-row-set sum;
// adding lane^16's value gives the full 32-lane sum (exact f32). EXEC must be all ones:
// called unconditionally by full 256-thread blocks.
__device__ __forceinline__ float wave_sum(float p) {
#ifdef HAVE_WMMA_X4
  v2f a;  a[0] = p;    a[1] = 0.0f;
  v2f o;  o[0] = 1.0f; o[1] = 1.0f;
  v8f c = {0.f, 0.f, 0.f, 0.f, 0.f, 0.f, 0.f, 0.f};
  v8f d = __builtin_amdgcn_wmma_f32_16x16x4_f32(false, a, false, o, (short)0, c, false, false);
  float s = ((d[0] + d[1]) + (d[2] + d[3])) + ((d[4] + d[5]) + (d[6] + d[7]));
  s += __shfl_xor(s, 16, 32);
  return s;
#else
  #pragma unroll
  for (int off = 16; off > 0; off >>= 1) p += __shfl_xor(p, off, 32);
  return p;
#endif
}

__device__ __forceinline__ void block_accum(float v, float* dst, float* red8) {
  float s = wave_sum(v);
  const int lane = threadIdx.x & 31, w = threadIdx.x >> 5;
  if (lane == 0) red8[w] = s;
  __syncthreads();
  if (threadIdx.x == 0) {
    float t = 0.f;
    const int nw = (blockDim.x + 31) >> 5;
    for (int i = 0; i < nw; ++i) t += red8[i];
    atomicAdd(dst, t);
  }
  __syncthreads();  // allow red8 reuse
}

// focal(x, y) with gamma=2:
//   z = y ? x : -x; p_t = sigmoid(z)
//   focal = alpha_f * sigmoid(-z)^2 * softplus(-z)   (== -alpha_f*(1-p_t)^2*log(p_t))
// 3 transcendentals per element: v_exp_f32, v_log_f32, v_rcp_f32.
__device__ __forceinline__ float focal_term(float x, bool pos) {
  const float u  = pos ? -x : x;                       // u = -z
  const float e  = __expf(-fabsf(u));                  // exp(-|u|), in (0,1]
  const float sp = fmaxf(u, 0.f) + __logf(1.f + e);    // softplus(u) = -log(p_t)
  const float r  = frcp(1.f + e);
  const float sg = (u >= 0.f) ? r : e * r;             // sigmoid(u) = 1 - p_t
  const float af = pos ? ALPHA_ : (1.f - ALPHA_);
  return af * sg * sg * sp;
}

// ---------------- kernel 1: focal cls loss, per (b, level, j-tile, channel-chunk) ----------------
// cls layout [B,C,h,w]; lane owns 4 contiguous j, loops 20 channels (stride HW floats).
// Channel rows are double-buffered through LDS via async b128 loads (ASYNCcnt).
__global__ void __launch_bounds__(256) fcos_focal_kernel(
    const float* __restrict__ cls, const int* __restrict__ tgt,
    float* __restrict__ ws, int HW, int s_off) {
#ifdef HAVE_ASYNC_LDS
  __shared__ float lbuf[2][TILE_J];
#endif
  __shared__ float red8[8];
  const int tid = threadIdx.x;
  const int b   = blockIdx.y;
  const int c0  = blockIdx.z * CCHUNK;
  const int jb  = blockIdx.x * TILE_J + tid * 4;   // HW % 4 == 0 at every level
  const bool valid = jb < HW;
  const int jc = valid ? jb : 0;                   // clamp: loads stay in-range, masked below

  const int4 tq = *(const int4*)(tgt + (size_t)b * S_ + s_off + jc);   // class ids 0..80
  const float* base = cls + ((size_t)(b * C_ + c0)) * HW + jc;

  float acc = 0.f;
#ifdef HAVE_ASYNC_LDS
  float* lp0 = &lbuf[0][tid * 4];
  float* lp1 = &lbuf[1][tid * 4];
  async_b128(base, lp0);                           // prime pipeline
  int cur = 0;
  for (int k = 0; k < CCHUNK; ++k) {
    if (k + 1 < CCHUNK) {
      async_b128(base + (size_t)(k + 1) * HW, cur ? lp0 : lp1);
      if (k + 2 < CCHUNK) __builtin_prefetch(base + (size_t)(k + 2) * HW, 0, 3);
      WAIT_ASYNC(1);                               // current buffer complete
    } else {
      WAIT_ASYNC(0);
    }
    const float4 x = *(const float4*)(cur ? lp1 : lp0);
    const int id = c0 + k + 1;                     // onehot classes are 1..80
    const float f = focal_term(x.x, tq.x == id) + focal_term(x.y, tq.y == id)
                  + focal_term(x.z, tq.z == id) + focal_term(x.w, tq.w == id);
    acc += valid ? f : 0.f;
    cur ^= 1;
  }
#else
  for (int k = 0; k < CCHUNK; ++k) {
    const float4 x = *(const float4*)(base + (size_t)k * HW);
    const int id = c0 + k + 1;
    const float f = focal_term(x.x, tq.x == id) + focal_term(x.y, tq.y == id)
                  + focal_term(x.z, tq.z == id) + focal_term(x.w, tq.w == id);
    acc += valid ? f : 0.f;
  }
#endif
  block_accum(acc, ws + b, red8);                  // ws[0..15] = per-batch cls sums
}

// ---------------- kernel 2: centerness BCE + GIoU + positive count ----------------
__global__ void __launch_bounds__(256) fcos_cntreg_kernel(
    const float* __restrict__ cnt, const float* __restrict__ reg,
    const float* __restrict__ cnt_t, const float* __restrict__ reg_t,
    float* __restrict__ ws, int HW, int s_off) {
  __shared__ float red8[8];
  const int tid = threadIdx.x;
  const int b   = blockIdx.y;
  const int j   = blockIdx.x * 256 + tid;
  const bool valid = j < HW;
  const int jc = valid ? j : 0;
  const int s  = s_off + jc;

  const float x  = cnt[(size_t)b * HW + jc];                       // [B,1,h,w]
  const float tg = cnt_t[(size_t)b * S_ + s];                      // [B,S,1]
  const float m  = (valid && tg > -1.f) ? 1.f : 0.f;

  // BCE with logits
  const float bce = fmaxf(x, 0.f) - x * tg + __logf(1.f + __expf(-fabsf(x)));

  // GIoU: preds [B,4,h,w] channel-strided, targets [B,S,4] as one float4
  const float pl = reg[((size_t)(b * 4 + 0)) * HW + jc];
  const float pt = reg[((size_t)(b * 4 + 1)) * HW + jc];
  const float pr = reg[((size_t)(b * 4 + 2)) * HW + jc];
  const float pb = reg[((size_t)(b * 4 + 3)) * HW + jc];
  const float4 tv = ((const float4*)reg_t)[(size_t)b * S_ + s];    // l,t,r,b

  const float wmin = fmaxf(fminf(pr, tv.z) + fminf(pl, tv.x), 0.f);
  const float hmin = fmaxf(fminf(pb, tv.w) + fminf(pt, tv.y), 0.f);
  const float overlap = wmin * hmin;
  const float area1 = (pr + pl) * (pb + pt);
  const float area2 = (tv.z + tv.x) * (tv.w + tv.y);
  const float uni = area1 + area2 - overlap;
  const float iou = overlap * frcp(uni);
  const float wmax = fmaxf(fmaxf(pr, tv.z) + fmaxf(pl, tv.x), 0.f);
  const float hmax = fmaxf(fmaxf(pb, tv.w) + fmaxf(pt, tv.y), 0.f);
  const float g = wmax * hmax;
  const float giou = iou - (g - uni) * frcp(fmaxf(g, 1e-10f));

  block_accum(bce * m,          ws + 16 + b, red8);   // cnt sums
  block_accum((1.f - giou) * m, ws + 32 + b, red8);   // reg sums
  block_accum(m,                ws + 48 + b, red8);   // positive counts
}

// ---------------- helpers: zero workspace / finalize ----------------
__global__ void fcos_zero_ws(float* ws) {
  if (threadIdx.x < 64) ws[threadIdx.x] = 0.f;
}

__global__ void fcos_finalize(const float* __restrict__ ws, float* __restrict__ out) {
  if (threadIdx.x == 0) {
    float cl = 0.f, cn = 0.f, rg = 0.f;
    for (int b = 0; b < B_; ++b) {
      const float inv = 1.f / fmaxf(ws[48 + b], 1.f);   // clip(num_pos, 1)
      cl += ws[b] * inv;
      cn += ws[16 + b] * inv;
      rg += ws[32 + b] * inv;
    }
    cl *= (1.f / B_); cn *= (1.f / B_); rg *= (1.f / B_);
    out[0] = cl; out[1] = cn; out[2] = rg; out[3] = cl + cn + rg;
  }
}

// ---------------- launch ----------------
extern "C" void kernel_launch(void* const* d_in, const int* in_sizes, int n_in,
                              void* d_out, int out_size, void* d_ws, size_t ws_size,
                              hipStream_t stream) {
  static const int HWs[5]  = {12800, 3200, 800, 208, 56};
  static const int offs[5] = {0, 12800, 16000, 16800, 17008};
  float* ws = (float*)d_ws;                       // 64 floats: cls|cnt|reg|pos per batch
  const float* cnt_t = (const float*)d_in[15];
  const float* reg_t = (const float*)d_in[16];
  const int*   cls_t = (const int*)d_in[17];

  fcos_zero_ws<<<1, 64, 0, stream>>>(ws);
  for (int i = 0; i < 5; ++i) {
    const float* cls = (const float*)d_in[i * 3 + 0];
    const float* cnt = (const float*)d_in[i * 3 + 1];
    const float* reg = (const float*)d_in[i * 3 + 2];
    dim3 gf((HWs[i] + TILE_J - 1) / TILE_J, B_, CSPLIT);
    fcos_focal_kernel<<<gf, 256, 0, stream>>>(cls, cls_t, ws, HWs[i], offs[i]);
    dim3 gc((HWs[i] + 255) / 256, B_);
    fcos_cntreg_kernel<<<gc, 256, 0, stream>>>(cnt, reg, cnt_t, reg_t, ws, HWs[i], offs[i]);
  }
  fcos_finalize<<<1, 32, 0, stream>>>(ws, (float*)d_out);
}